// NN_35562329211597
// MI455X (gfx1250) — compile-verified
//
#include <hip/hip_runtime.h>
#include <hip/hip_bf16.h>

typedef __attribute__((ext_vector_type(2))) float v2f;
typedef __attribute__((ext_vector_type(8))) float v8f;

#define BM 128
#define BN 128
#define BKC 32
#define LPAD 34
#define SPLIT 32

// ---------------- BN stats (deterministic two-stage column reduction) ----------------
__global__ void colstats_partial(const float* __restrict__ X, int M, int D,
                                 float* __restrict__ partial) {
    int d = blockIdx.x * 256 + threadIdx.x;
    int s = blockIdx.y;
    int rows = M / SPLIT;
    int r0 = s * rows;
    float s1 = 0.f, s2 = 0.f;
    for (int m = 0; m < rows; ++m) {
        float v = X[(size_t)(r0 + m) * D + d];
        s1 += v; s2 += v * v;
    }
    partial[(size_t)s * 2 * D + d]     = s1;
    partial[(size_t)s * 2 * D + D + d] = s2;
}

__global__ void finalize_ac(const float* __restrict__ partial,
                            const float* __restrict__ g, const float* __restrict__ be,
                            float* __restrict__ a, float* __restrict__ c, int D, int M) {
    int d = blockIdx.x * 256 + threadIdx.x;
    float s1 = 0.f, s2 = 0.f;
    for (int s = 0; s < SPLIT; ++s) {
        s1 += partial[(size_t)s * 2 * D + d];
        s2 += partial[(size_t)s * 2 * D + D + d];
    }
    float mu  = s1 / (float)M;
    float var = fmaxf(s2 / (float)M - mu * mu, 0.0f);
    float av  = g[d] * rsqrtf(var + 1e-5f);
    a[d] = av;
    c[d] = be[d] - mu * av;
}

// ---------------- fold BN affine into weights: W'=a*W, b'=b+c@W ----------------
__global__ void fold_w(const float* __restrict__ W, const float* __restrict__ b,
                       const float* __restrict__ a, const float* __restrict__ c,
                       float* __restrict__ Wf, float* __restrict__ bf,
                       int Din, int Dout) {
    int f = blockIdx.x * 256 + threadIdx.x;
    float acc = 0.f;
    for (int d = 0; d < Din; ++d) {
        float w = W[(size_t)d * Dout + f];
        Wf[(size_t)d * Dout + f] = a[d] * w;
        acc += c[d] * w;
    }
    bf[f] = b[f] + acc;
}

// ---------------- fused GEMM + bias + tanh via fp32 WMMA, 2x2 reg blocking ----------------
__global__ __launch_bounds__(512)
void gemm_bias_tanh(const float* __restrict__ X, const float* __restrict__ Wf,
                    const float* __restrict__ bf, float* __restrict__ Y,
                    int M, int N, int K) {
    __shared__ float As[BM][LPAD];   // As[m][k]
    __shared__ float Bs[BN][LPAD];   // Bs[n][k] (transposed stage)
    int tid  = threadIdx.x;
    int lane = tid & 31;
    int w    = tid >> 5;
    int wy   = w >> 2, wx = w & 3;
    int l15  = lane & 15, hi = lane >> 4;
    int m0 = blockIdx.y * BM;
    int n0 = blockIdx.x * BN;

    v8f acc[2][2];
#pragma unroll
    for (int s = 0; s < 2; ++s)
#pragma unroll
        for (int t = 0; t < 2; ++t)
            acc[s][t] = (v8f){0.f,0.f,0.f,0.f,0.f,0.f,0.f,0.f};

    for (int k0 = 0; k0 < K; k0 += BKC) {
        __syncthreads();
#pragma unroll
        for (int i = 0; i < 8; ++i) {                 // A: 128x32
            int e = tid + i * 512;
            int m = e >> 5, k = e & 31;
            As[m][k] = X[(size_t)(m0 + m) * K + k0 + k];
        }
#pragma unroll
        for (int i = 0; i < 8; ++i) {                 // B: 32x128 -> transposed
            int e = tid + i * 512;
            int k = e >> 7, n = e & 127;
            Bs[n][k] = Wf[(size_t)(k0 + k) * N + n0 + n];
        }
        __syncthreads();
#pragma unroll
        for (int kk = 0; kk < BKC; kk += 4) {
            int kb = kk + hi * 2;
            v2f a[2], b[2];
#pragma unroll
            for (int s = 0; s < 2; ++s) {
                int mr = wy * 32 + s * 16 + l15;
                a[s].x = As[mr][kb]; a[s].y = As[mr][kb + 1];
                int nr = wx * 32 + s * 16 + l15;
                b[s].x = Bs[nr][kb]; b[s].y = Bs[nr][kb + 1];
            }
#pragma unroll
            for (int s = 0; s < 2; ++s)
#pragma unroll
                for (int t = 0; t < 2; ++t)
                    acc[s][t] = __builtin_amdgcn_wmma_f32_16x16x4_f32(
                        false, a[s], false, b[t], (short)0, acc[s][t], false, false);
        }
    }
#pragma unroll
    for (int t = 0; t < 2; ++t) {
        int n = n0 + wx * 32 + t * 16 + l15;
        float bias = bf[n];
#pragma unroll
        for (int s = 0; s < 2; ++s)
#pragma unroll
            for (int r = 0; r < 8; ++r) {
                int m = m0 + wy * 32 + s * 16 + hi * 8 + r;
                Y[(size_t)m * N + n] = tanhf(acc[s][t][r] + bias);
            }
    }
}

// ---------------- per-row squared norm of K ----------------
__global__ void rowsq(const float* __restrict__ Km, float* __restrict__ k2,
                      int NN, int D) {
    int w = (blockIdx.x * blockDim.x + threadIdx.x) >> 5;
    int lane = threadIdx.x & 31;
    if (w >= NN) return;
    float s = 0.f;
    for (int d = lane; d < D; d += 32) { float v = Km[(size_t)w * D + d]; s += v * v; }
    for (int off = 16; off; off >>= 1) s += __shfl_xor(s, off, 32);
    if (lane == 0) k2[w] = s;
}

// ---------------- pass 1: row max of sim' = 2 Q.K^T - k2 ----------------
__global__ __launch_bounds__(512)
void sim_rowmax(const float* __restrict__ Q, const float* __restrict__ Km,
                const float* __restrict__ k2, float* __restrict__ rowmax,
                int NN, int D) {
    __shared__ float As[BM][LPAD];
    __shared__ float Bs[BN][LPAD];
    __shared__ float red[4][BM];
    int tid  = threadIdx.x;
    int lane = tid & 31;
    int w    = tid >> 5;
    int wy   = w >> 2, wx = w & 3;
    int l15  = lane & 15, hi = lane >> 4;
    int m0 = blockIdx.x * BM;

    float vmax[2][8];
#pragma unroll
    for (int s = 0; s < 2; ++s)
#pragma unroll
        for (int r = 0; r < 8; ++r) vmax[s][r] = -3.0e38f;

    for (int n0 = 0; n0 < NN; n0 += BN) {
        v8f acc[2][2];
#pragma unroll
        for (int s = 0; s < 2; ++s)
#pragma unroll
            for (int t = 0; t < 2; ++t)
                acc[s][t] = (v8f){0.f,0.f,0.f,0.f,0.f,0.f,0.f,0.f};

        for (int k0 = 0; k0 < D; k0 += BKC) {
            __syncthreads();
#pragma unroll
            for (int i = 0; i < 8; ++i) {
                int e = tid + i * 512;
                int m = e >> 5, k = e & 31;
                As[m][k] = Q[(size_t)(m0 + m) * D + k0 + k];
            }
#pragma unroll
            for (int i = 0; i < 8; ++i) {
                int e = tid + i * 512;
                int n = e >> 5, k = e & 31;
                Bs[n][k] = Km[(size_t)(n0 + n) * D + k0 + k];
            }
            __syncthreads();
#pragma unroll
            for (int kk = 0; kk < BKC; kk += 4) {
                int kb = kk + hi * 2;
                v2f a[2], b[2];
#pragma unroll
                for (int s = 0; s < 2; ++s) {
                    int mr = wy * 32 + s * 16 + l15;
                    a[s].x = As[mr][kb]; a[s].y = As[mr][kb + 1];
                    int nr = wx * 32 + s * 16 + l15;
                    b[s].x = Bs[nr][kb]; b[s].y = Bs[nr][kb + 1];
                }
#pragma unroll
                for (int s = 0; s < 2; ++s)
#pragma unroll
                    for (int t = 0; t < 2; ++t)
                        acc[s][t] = __builtin_amdgcn_wmma_f32_16x16x4_f32(
                            false, a[s], false, b[t], (short)0, acc[s][t], false, false);
            }
        }
        float k2v[2];
#pragma unroll
        for (int t = 0; t < 2; ++t) k2v[t] = k2[n0 + wx * 32 + t * 16 + l15];
#pragma unroll
        for (int s = 0; s < 2; ++s)
#pragma unroll
            for (int t = 0; t < 2; ++t)
#pragma unroll
                for (int r = 0; r < 8; ++r)
                    vmax[s][r] = fmaxf(vmax[s][r], 2.0f * acc[s][t][r] - k2v[t]);
    }
#pragma unroll
    for (int s = 0; s < 2; ++s)
#pragma unroll
        for (int r = 0; r < 8; ++r)
            for (int off = 1; off < 16; off <<= 1)
                vmax[s][r] = fmaxf(vmax[s][r], __shfl_xor(vmax[s][r], off, 32));
    if (l15 == 0) {
#pragma unroll
        for (int s = 0; s < 2; ++s)
#pragma unroll
            for (int r = 0; r < 8; ++r)
                red[wx][wy * 32 + s * 16 + hi * 8 + r] = vmax[s][r];
    }
    __syncthreads();
    if (tid < BM) {
        float m = fmaxf(fmaxf(red[0][tid], red[1][tid]),
                        fmaxf(red[2][tid], red[3][tid]));
        rowmax[m0 + tid] = m;
    }
}

// ---------------- pass 2: streaming softmax @ y ----------------
__global__ __launch_bounds__(512)
void sim_out(const float* __restrict__ Q, const float* __restrict__ Km,
             const float* __restrict__ k2, const float* __restrict__ y,
             const float* __restrict__ rowmax, float* __restrict__ out,
             int NN, int D) {
    __shared__ float As[BM][LPAD];
    __shared__ float Bs[BN][LPAD];
    __shared__ float yls[BN];
    __shared__ float redN[4][BM];
    __shared__ float redD[4][BM];
    int tid  = threadIdx.x;
    int lane = tid & 31;
    int w    = tid >> 5;
    int wy   = w >> 2, wx = w & 3;
    int l15  = lane & 15, hi = lane >> 4;
    int m0 = blockIdx.x * BM;

    float rm[2][8], num[2][8], den[2][8];
#pragma unroll
    for (int s = 0; s < 2; ++s)
#pragma unroll
        for (int r = 0; r < 8; ++r) {
            rm[s][r]  = rowmax[m0 + wy * 32 + s * 16 + hi * 8 + r];
            num[s][r] = 0.f; den[s][r] = 0.f;
        }

    for (int n0 = 0; n0 < NN; n0 += BN) {
        v8f acc[2][2];
#pragma unroll
        for (int s = 0; s < 2; ++s)
#pragma unroll
            for (int t = 0; t < 2; ++t)
                acc[s][t] = (v8f){0.f,0.f,0.f,0.f,0.f,0.f,0.f,0.f};

        for (int k0 = 0; k0 < D; k0 += BKC) {
            __syncthreads();
#pragma unroll
            for (int i = 0; i < 8; ++i) {
                int e = tid + i * 512;
                int m = e >> 5, k = e & 31;
                As[m][k] = Q[(size_t)(m0 + m) * D + k0 + k];
            }
#pragma unroll
            for (int i = 0; i < 8; ++i) {
                int e = tid + i * 512;
                int n = e >> 5, k = e & 31;
                Bs[n][k] = Km[(size_t)(n0 + n) * D + k0 + k];
            }
            if (k0 == 0 && tid < BN) yls[tid] = y[n0 + tid];
            __syncthreads();
#pragma unroll
            for (int kk = 0; kk < BKC; kk += 4) {
                int kb = kk + hi * 2;
                v2f a[2], b[2];
#pragma unroll
                for (int s = 0; s < 2; ++s) {
                    int mr = wy * 32 + s * 16 + l15;
                    a[s].x = As[mr][kb]; a[s].y = As[mr][kb + 1];
                    int nr = wx * 32 + s * 16 + l15;
                    b[s].x = Bs[nr][kb]; b[s].y = Bs[nr][kb + 1];
                }
#pragma unroll
                for (int s = 0; s < 2; ++s)
#pragma unroll
                    for (int t = 0; t < 2; ++t)
                        acc[s][t] = __builtin_amdgcn_wmma_f32_16x16x4_f32(
                            false, a[s], false, b[t], (short)0, acc[s][t], false, false);
            }
        }
        float k2v[2], yv[2];
#pragma unroll
        for (int t = 0; t < 2; ++t) {
            k2v[t] = k2[n0 + wx * 32 + t * 16 + l15];
            yv[t]  = yls[wx * 32 + t * 16 + l15];
        }
#pragma unroll
        for (int s = 0; s < 2; ++s)
#pragma unroll
            for (int t = 0; t < 2; ++t)
#pragma unroll
                for (int r = 0; r < 8; ++r) {
                    float sv = 2.0f * acc[s][t][r] - k2v[t];
                    float e  = __expf(sv - rm[s][r]);
                    den[s][r] += e;
                    num[s][r] += e * yv[t];
                }
    }
#pragma unroll
    for (int s = 0; s < 2; ++s)
#pragma unroll
        for (int r = 0; r < 8; ++r)
            for (int off = 1; off < 16; off <<= 1) {
                num[s][r] += __shfl_xor(num[s][r], off, 32);
                den[s][r] += __shfl_xor(den[s][r], off, 32);
            }
    if (l15 == 0) {
#pragma unroll
        for (int s = 0; s < 2; ++s)
#pragma unroll
            for (int r = 0; r < 8; ++r) {
                redN[wx][wy * 32 + s * 16 + hi * 8 + r] = num[s][r];
                redD[wx][wy * 32 + s * 16 + hi * 8 + r] = den[s][r];
            }
    }
    __syncthreads();
    if (tid < BM) {
        float nm = redN[0][tid] + redN[1][tid] + redN[2][tid] + redN[3][tid];
        float dn = redD[0][tid] + redD[1][tid] + redD[2][tid] + redD[3][tid];
        out[m0 + tid] = fminf(fmaxf(nm / dn, 0.0f), 1.0f);
    }
}

// ---------------- host-side orchestration ----------------
extern "C" void kernel_launch(void* const* d_in, const int* in_sizes, int n_in,
                              void* d_out, int out_size, void* d_ws, size_t ws_size,
                              hipStream_t stream) {
    (void)in_sizes; (void)n_in; (void)out_size; (void)ws_size;

    const float* x  = (const float*)d_in[0];   // [4096,1024]
    const float* xn = (const float*)d_in[1];   // [8192,1024]
    const float* yn = (const float*)d_in[2];   // [8192]
    const float* Wl[3]  = {(const float*)d_in[3], (const float*)d_in[7],  (const float*)d_in[11]};
    const float* bl[3]  = {(const float*)d_in[4], (const float*)d_in[8],  (const float*)d_in[12]};
    const float* gl[3]  = {(const float*)d_in[5], (const float*)d_in[9],  (const float*)d_in[13]};
    const float* bel[3] = {(const float*)d_in[6], (const float*)d_in[10], (const float*)d_in[14]};

    const int B = 4096, N = 8192;
    const int dims[4] = {1024, 1024, 512, 256};

    float* ws = (float*)d_ws;
    size_t off = 0;
    float* T1   = ws + off; off += (size_t)8192 * 1024;   // layer-1 activations
    float* T2   = ws + off; off += (size_t)8192 * 512;    // layer-2 activations
    float* Qb   = ws + off; off += (size_t)4096 * 256;    // q
    float* Kb   = ws + off; off += (size_t)8192 * 256;    // k
    float* Wf   = ws + off; off += (size_t)1024 * 1024;   // folded weights
    float* bfv  = ws + off; off += 1024;                  // folded bias
    float* av   = ws + off; off += 1024;                  // BN scale
    float* cv   = ws + off; off += 1024;                  // BN shift
    float* part = ws + off; off += (size_t)SPLIT * 2 * 1024; // stats partials
    float* k2v  = ws + off; off += 8192;                  // ||k_j||^2
    float* rmax = ws + off; off += 4096;                  // row max

    auto run_layer = [&](const float* Xin, int Mrows, int Din, int Dout,
                         const float* Wp, const float* bp, const float* gp,
                         const float* bep, float* Yout) {
        dim3 gs(Din / 256, SPLIT);
        colstats_partial<<<gs, 256, 0, stream>>>(Xin, Mrows, Din, part);
        finalize_ac<<<Din / 256, 256, 0, stream>>>(part, gp, bep, av, cv, Din, Mrows);
        fold_w<<<Dout / 256, 256, 0, stream>>>(Wp, bp, av, cv, Wf, bfv, Din, Dout);
        dim3 gg(Dout / BN, Mrows / BM);
        gemm_bias_tanh<<<gg, 512, 0, stream>>>(Xin, Wf, bfv, Yout, Mrows, Dout, Din);
    };

    // q = trunk(x)
    run_layer(x,  B, dims[0], dims[1], Wl[0], bl[0], gl[0], bel[0], T1);
    run_layer(T1, B, dims[1], dims[2], Wl[1], bl[1], gl[1], bel[1], T2);
    run_layer(T2, B, dims[2], dims[3], Wl[2], bl[2], gl[2], bel[2], Qb);
    // k = trunk(x_n)  (its own batch stats)
    run_layer(xn, N, dims[0], dims[1], Wl[0], bl[0], gl[0], bel[0], T1);
    run_layer(T1, N, dims[1], dims[2], Wl[1], bl[1], gl[1], bel[1], T2);
    run_layer(T2, N, dims[2], dims[3], Wl[2], bl[2], gl[2], bel[2], Kb);

    rowsq<<<(N * 32) / 256, 256, 0, stream>>>(Kb, k2v, N, 256);
    sim_rowmax<<<B / BM, 512, 0, stream>>>(Qb, Kb, k2v, rmax, N, 256);
    sim_out<<<B / BM, 512, 0, stream>>>(Qb, Kb, k2v, yn, rmax, (float*)d_out, N, 256);
}